// SpatialDerivatives_33320356282811
// MI455X (gfx1250) — compile-verified
//
#include <hip/hip_runtime.h>

typedef float v2f __attribute__((ext_vector_type(2)));
typedef float v8f __attribute__((ext_vector_type(8)));

#define EPSF 1e-8f

// Fixed reference shapes: u,v,w = [NB, NY, NX, NZ] f32, z innermost.
constexpr int NB = 2;
constexpr int NY = 721;
constexpr int NX = 1440;
constexpr int NZ = 13;
constexpr int WAVES_PER_BLOCK = 2;              // blockDim = 64
constexpr int XT_PER_ROW = NX / 16;             // 90 x-tiles per row (exact)
constexpr int WTILE = 16 * NZ;                  // 208 floats = 832 contiguous bytes

// gfx1250 async global->LDS staging, if the toolchain exposes the builtins.
#if defined(__has_builtin)
#if __has_builtin(__builtin_amdgcn_global_load_async_to_lds_b32) && \
    __has_builtin(__builtin_amdgcn_s_wait_asynccnt)
#define USE_ASYNC_LDS 1
#endif
#endif
#ifndef USE_ASYNC_LDS
#define USE_ASYNC_LDS 0
#endif

#if USE_ASYNC_LDS
// builtin signature (from hipcc diagnostic): param0 = addrspace(1) int*  ("__device__ int *")
typedef __attribute__((address_space(1))) int g_i32;
typedef __attribute__((address_space(3))) int l_i32;
#endif

// One wave = 16 consecutive columns of one (b,y) row.
//   horizontal (du/dx periodic + dv/dy one-sided) -> WMMA accumulator C
//   vertical   (dw/dsigma, banded 13x13 matrix)   -> 4x v_wmma_f32_16x16x4_f32
__global__ __launch_bounds__(64) void spatial_div_wmma(
    const float* __restrict__ u,
    const float* __restrict__ v,
    const float* __restrict__ w,
    const float* __restrict__ dsig,
    const float* __restrict__ dx,
    const float* __restrict__ dy,
    float* __restrict__ out)
{
    const int lane = threadIdx.x & 31;
    const int wave = threadIdx.x >> 5;
    const int half = lane >> 4;                 // 0: z rows 0-7, 1: z rows 8-15
    const int m    = lane & 15;                 // A: matrix row (output z); B/C/D: column id

    const int row = blockIdx.y;                 // bb*NY + y   (0 .. NB*NY-1)
    const int xt  = blockIdx.x * WAVES_PER_BLOCK + wave;   // 0 .. 89 (exact)
    const int x0  = xt * 16;
    const int x   = x0 + m;

    const int bb = (row >= NY) ? 1 : 0;
    const int y  = row - bb * NY;

    // periodic x neighbors
    const int xp = (x == NX - 1) ? 0 : x + 1;
    const int xm = (x == 0) ? (NX - 1) : x - 1;
    // one-sided y boundaries (no 1/2 factor at the ends, matching reference)
    int row0, row1; float fy;
    if (y == 0)           { row0 = row;     row1 = row + 1; fy = 1.0f; }
    else if (y == NY - 1) { row0 = row - 1; row1 = row;     fy = 1.0f; }
    else                  { row0 = row - 1; row1 = row + 1; fy = 0.5f; }

    const unsigned rowBase = (unsigned)row * NX;            // column index of (bb,y,0)
    const float* up = u + (size_t)(rowBase + xp) * NZ;
    const float* um = u + (size_t)(rowBase + xm) * NZ;
    const float* vp = v + (size_t)((unsigned)row1 * NX + x) * NZ;
    const float* vm = v + (size_t)((unsigned)row0 * NX + x) * NZ;

    const float invdx = 0.5f / (dx[(unsigned)y * NX + x] + EPSF);
    const float invdy = fy   / (dy[(unsigned)y * NX + x] + EPSF);

    // ---- w source for the B matrix: async-stage 832B tile to LDS if available ----
#if USE_ASYNC_LDS
    __shared__ float wstage[WAVES_PER_BLOCK][WTILE];
    {
        const float* gsrc = w + (size_t)(rowBase + x0) * NZ;   // 208 contiguous floats
#pragma unroll
        for (int i = 0; i < (WTILE + 31) / 32; ++i) {
            const int idx = i * 32 + lane;
            if (idx < WTILE) {
                __builtin_amdgcn_global_load_async_to_lds_b32(
                    (g_i32*)(gsrc + idx),
                    (l_i32*)&wstage[wave][idx],
                    /*offset=*/0, /*cpol=*/0);
            }
        }
        __builtin_amdgcn_s_wait_asynccnt(0);   // wave-private data: no barrier needed
    }
    const float* wc = &wstage[wave][m * NZ];   // lane's column in LDS -> ds_load
#else
    const float* wc = w + (size_t)(rowBase + x) * NZ;
#endif

    // ---- horizontal stencil terms into the accumulator
    //      (C/D layout: lane L owns z = 8*(L/16)+t, col = L%16, t = VGPR index) ----
    const int zbase = half * 8;
    v8f acc;
#pragma unroll
    for (int t = 0; t < 8; ++t) {
        const int z = zbase + t;
        float s = 0.0f;
        if (z < NZ) {
            s = (up[z] - um[z]) * invdx + (vp[z] - vm[z]) * invdy;
        }
        acc[t] = s;
    }

    // ---- banded vertical-derivative matrix row for this lane (A matrix) ----
    //  row 0:    [-r, +r, 0, ...]              r = 1/(ds0 + eps)
    //  row mid:  [-r @ z-1, 0, +r @ z+1]       r = 1/(ds[z-1]+ds[z] + eps)
    //  row NZ-1: [..., -r, +r]                 r = 1/(ds[NZ-1] + eps)
    //  rows >= NZ: zero (padding)
    float r = 0.0f;
    if (m == 0)            r = 1.0f / (dsig[0] + EPSF);
    else if (m == NZ - 1)  r = 1.0f / (dsig[NZ - 1] + EPSF);
    else if (m < NZ - 1)   r = 1.0f / (dsig[m - 1] + dsig[m] + EPSF);
    const int lo = (m == 0) ? 0 : m - 1;
    const int hi = (m >= NZ - 1) ? (NZ - 1) : m + 1;

    // ---- 4 chained f32 WMMAs over K = 16 (NZ = 13, zero-padded) ----
    // A layout: lane L = row m=L%16, VGPR j holds local K = 2*(L/16)+j
    // B layout: lane L = col  L%16,  VGPR j holds local K = 2*(L/16)+j
#pragma unroll
    for (int j = 0; j < 4; ++j) {
        const int k0 = 4 * j + 2 * half;
        v2f a, b2;
        a.x = r * ((float)(k0     == hi) - (float)(k0     == lo));
        a.y = r * ((float)(k0 + 1 == hi) - (float)(k0 + 1 == lo));
        b2.x = (k0     < NZ) ? wc[k0]     : 0.0f;
        b2.y = (k0 + 1 < NZ) ? wc[k0 + 1] : 0.0f;
        acc = __builtin_amdgcn_wmma_f32_16x16x4_f32(
            /*neg_a=*/false, a, /*neg_b=*/false, b2,
            /*c_mod=*/(short)0, acc, /*reuse_a=*/false, /*reuse_b=*/false);
    }

    // ---- store D: lane L writes its contiguous z range of its column ----
    float* oc = out + (size_t)(rowBase + x) * NZ;
#pragma unroll
    for (int t = 0; t < 8; ++t) {
        const int z = zbase + t;
        if (z < NZ) oc[z] = acc[t];
    }

    // gfx1250 global_prefetch_b8: warm L2 with the next row's w tile (speculative)
    __builtin_prefetch((const void*)(w + (size_t)(rowBase + x0) * NZ + (size_t)NX * NZ), 0, 1);
}

extern "C" void kernel_launch(void* const* d_in, const int* in_sizes, int n_in,
                              void* d_out, int out_size, void* d_ws, size_t ws_size,
                              hipStream_t stream) {
    const float* u  = (const float*)d_in[0];
    const float* v  = (const float*)d_in[1];
    const float* w  = (const float*)d_in[2];
    const float* ds = (const float*)d_in[3];
    const float* dx = (const float*)d_in[4];
    const float* dy = (const float*)d_in[5];
    float* out = (float*)d_out;

    // grid: x = 90 x-tiles / 2 waves per block = 45 blocks, y = NB*NY rows = 1442
    dim3 grid(XT_PER_ROW / WAVES_PER_BLOCK, NB * NY);
    dim3 block(WAVES_PER_BLOCK * 32);

    spatial_div_wmma<<<grid, block, 0, stream>>>(u, v, w, ds, dx, dy, out);
}